// PCSNNTorchActive_9990093930949
// MI455X (gfx1250) — compile-verified
//
#include <hip/hip_runtime.h>

typedef __attribute__((ext_vector_type(2))) float v2f;
typedef __attribute__((ext_vector_type(8))) float v8f;

#define NR 300
#define BB 4
#define NN 512
#define REG_ELEMS (BB * NN)          // 2048
#define TOT (NR * REG_ELEMS)         // 614400
#define WTOT (NR * NN * NN)          // 78643200
#define TAU 0.8f
#define ALPHA 0.05f
#define HEBB_LR 2e-5f
#define NOISE_SCALE 0.02f
#define SCAN_WGS 32
#define FPART_BLOCKS 1024
#define XPITCH 514                   // 512 + 2: kills LDS bank conflicts on column reads

// ---------------------------------------------------------------- init
__global__ __launch_bounds__(256) void pcsnn_init_kernel(
    const float* __restrict__ vs0, const float* __restrict__ noise,
    float* __restrict__ vs_init, unsigned* __restrict__ sync) {
  if (blockIdx.x == 0 && threadIdx.x < 2) sync[threadIdx.x] = 0u;  // count, gen
  int idx = blockIdx.x * 256 + threadIdx.x;
  if (idx < TOT) vs_init[idx] = vs0[idx] + NOISE_SCALE * noise[idx];
}

// ------------------------------------------------- base pre-activation
// P[i,b,c] = inp[i,b,c] + sum_j A[j,i] * vs_init[j,b,c]
__global__ __launch_bounds__(256) void pcsnn_pre_kernel(
    const float* __restrict__ A, const float* __restrict__ vs_init,
    const float* __restrict__ inp, float* __restrict__ P) {
  int idx = blockIdx.x * 256 + threadIdx.x;
  if (idx >= TOT) return;
  int i = idx / REG_ELEMS;
  int bc = idx % REG_ELEMS;
  float acc = inp[idx];
  for (int j = 0; j < NR; ++j)
    acc += A[j * NR + i] * vs_init[j * REG_ELEMS + bc];
  P[idx] = acc;
}

// ------------------------------------------------------ grid-wide sync
__device__ __forceinline__ void grid_sync(unsigned* count, volatile unsigned* gen,
                                          unsigned nwg) {
  __syncthreads();
  if (threadIdx.x == 0) {
    __threadfence();
    unsigned g = *gen;
    if (atomicAdd(count, 1u) == nwg - 1u) {
      *count = 0u;
      __threadfence();
      atomicAdd((unsigned*)gen, 1u);
    } else {
      while (*gen == g) __builtin_amdgcn_s_sleep(1);
    }
  }
  __syncthreads();
}

// --------------------------------------------- sequential region scan
// 32 persistent WGs (one 16-col N-tile each), 8 waves split K=512.
// x[i] is staged in LDS padded to 16 rows so WMMA A-fragment reads are
// unconditional ds_load_b64 (no exec-masked global loads).
__global__ __launch_bounds__(256) void pcsnn_scan_kernel(
    const float* __restrict__ A, const float* __restrict__ W,
    const float* __restrict__ vb0, const float* __restrict__ va0,
    const float* __restrict__ P, const float* __restrict__ vs_init,
    float* __restrict__ xall, float* __restrict__ mism,
    float* __restrict__ delta, float* __restrict__ spikes,
    unsigned* __restrict__ sync) {
  __shared__ float xpad[16][XPITCH];   // rows 0-3: x[i]; rows 4-15: zero padding
  __shared__ float red[8][8][32];
  const int n0 = blockIdx.x * 16;
  const int t = threadIdx.x;
  const int wave = t >> 5;
  const int lane = t & 31;
  const int m = lane & 15;            // A-matrix row (M) / B-matrix column
  const int kh = (lane >> 4) << 1;    // lanes 0-15 -> K0/K1, lanes 16-31 -> K2/K3
  unsigned* count = sync;
  volatile unsigned* gen = sync + 1;

  // zero the M-padding rows once (never written again)
  for (int e = t; e < 12 * XPITCH; e += 256)
    xpad[4 + e / XPITCH][e % XPITCH] = 0.0f;

  for (int i = 0; i < NR; ++i) {
    grid_sync(count, gen, gridDim.x);  // delta[j<i] visible everywhere
    // ---- phase 1: x slice = P + Gauss-Seidel correction
    if (t < 64) {
      int b = t >> 4, c = n0 + (t & 15);
      int base = i * REG_ELEMS + b * NN + c;
      float acc = P[base];
      for (int j = 0; j < i; ++j)
        acc += A[j * NR + i] * delta[j * REG_ELEMS + b * NN + c];
      xall[base] = acc;
    } else if (i + 1 < NR) {
      // latency-hiding: pull next region's W column-slice toward the caches
      const float* Wn = W + (size_t)(i + 1) * NN * NN + n0;
      for (int r = t - 64; r < NN; r += 192)
        __builtin_prefetch(Wn + (size_t)r * NN, 0, 1);
    }
    grid_sync(count, gen, gridDim.x);  // full x[i] visible everywhere
    // stage x[i] (4x512) into LDS
    const float* xrow = xall + (size_t)i * REG_ELEMS;     // [4][512]
    for (int e = t; e < REG_ELEMS; e += 256)
      xpad[e >> 9][e & 511] = xrow[e];
    __syncthreads();
    // ---- phase 2: pred_b tile = x[4,512] @ W_i[512, n0:n0+16] via WMMA f32
    const float* Wr = W + (size_t)i * NN * NN;            // [512][512]
    v8f acc = {};
    for (int k = wave * 64; k < wave * 64 + 64; k += 4) {
      int kk = k + kh;                                    // even -> 8B aligned
      v2f a = *reinterpret_cast<const v2f*>(&xpad[m][kk]); // A: 16x4 (M padded)
      v2f b;
      b.x = Wr[(size_t)kk * NN + n0 + m];                 // B: 4x16
      b.y = Wr[(size_t)(kk + 1) * NN + n0 + m];
      acc = __builtin_amdgcn_wmma_f32_16x16x4_f32(
          false, a, false, b, (short)0, acc, false, false);
    }
#pragma unroll
    for (int r = 0; r < 8; ++r) red[wave][r][lane] = acc[r];
    __syncthreads();
    // ---- phase 3: cross-wave reduce + leaky integrate / spike
    if (t < 64) {
      int mm = t >> 4, c = t & 15;    // D layout: lanes 0-15, VGPR r -> (M=r, N=lane)
      float pred = 0.0f;
#pragma unroll
      for (int w = 0; w < 8; ++w) pred += red[w][mm][c];
      int base = i * REG_ELEMS + mm * NN + n0 + c;
      float vb = TAU * vb0[base] + (1.0f - TAU) * pred;
      float va = TAU * va0[base];
      float vsi = vs_init[base];
      float vs = TAU * vsi + (1.0f - TAU) * (vb - va);
      spikes[base] = (vs > 0.0f) ? 1.0f : 0.0f;
      mism[base] = vs - vb;
      delta[base] = vs - vsi;
    }
    __syncthreads();                  // red[]/xpad reused next region
  }
}

// ---------------------------------------- fused Hebbian weight update
// One 16x16 dW tile per wave: dW = rbar^T(16x4) x ebar(4x16), K=4 exact.
__global__ __launch_bounds__(256) void pcsnn_hebb_kernel(
    const float* __restrict__ W, const float* __restrict__ rbar0,
    const float* __restrict__ ebar0, const float* __restrict__ xall,
    const float* __restrict__ mism, float* __restrict__ Wout) {
  const int t = threadIdx.x, wave = t >> 5, lane = t & 31;
  const long tile = (long)blockIdx.x * 8 + wave;  // 300*32*32 tiles total
  const int i = (int)(tile >> 10);
  const int within = (int)(tile & 1023);
  const int k0 = (within >> 5) * 16;
  const int n0 = (within & 31) * 16;
  const int m = lane & 15;
  const int bh = (lane >> 4) << 1;    // batch pair for this lane half
  const long rb = (long)i * REG_ELEMS;
  v2f a, b;
  // rbar(b,k): A-matrix 16(k) x 4(b)      ebar(b,n): B-matrix 4(b) x 16(n)
  a.x = (1.0f - ALPHA) * rbar0[rb + bh * NN + k0 + m] + ALPHA * xall[rb + bh * NN + k0 + m];
  a.y = (1.0f - ALPHA) * rbar0[rb + (bh + 1) * NN + k0 + m] + ALPHA * xall[rb + (bh + 1) * NN + k0 + m];
  b.x = (1.0f - ALPHA) * ebar0[rb + bh * NN + n0 + m] + ALPHA * mism[rb + bh * NN + n0 + m];
  b.y = (1.0f - ALPHA) * ebar0[rb + (bh + 1) * NN + n0 + m] + ALPHA * mism[rb + (bh + 1) * NN + n0 + m];
  v8f c = {};
  c = __builtin_amdgcn_wmma_f32_16x16x4_f32(false, a, false, b, (short)0, c, false, false);
  const float s = HEBB_LR / (float)BB;
  const int rowadd = (lane >= 16) ? 8 : 0;
  const size_t wbase = (size_t)i * NN * NN;
#pragma unroll
  for (int r = 0; r < 8; ++r) {
    size_t off = wbase + (size_t)(k0 + r + rowadd) * NN + n0 + (lane & 15);
    Wout[off] = W[off] + s * c[r];
  }
}

// --------------------------------------------------- F = mean(mism^2)
__global__ __launch_bounds__(256) void pcsnn_fpart_kernel(
    const float* __restrict__ mism, float* __restrict__ partials) {
  __shared__ float s[256];
  float acc = 0.0f;
  for (int idx = blockIdx.x * 256 + threadIdx.x; idx < TOT; idx += gridDim.x * 256) {
    float v = mism[idx];
    acc += v * v;
  }
  s[threadIdx.x] = acc;
  __syncthreads();
  for (int k = 128; k > 0; k >>= 1) {
    if (threadIdx.x < k) s[threadIdx.x] += s[threadIdx.x + k];
    __syncthreads();
  }
  if (threadIdx.x == 0) partials[blockIdx.x] = s[0];
}

__global__ void pcsnn_ffinal_kernel(const float* __restrict__ partials,
                                    float* __restrict__ out) {
  if (threadIdx.x == 0 && blockIdx.x == 0) {
    float acc = 0.0f;
    for (int p = 0; p < FPART_BLOCKS; ++p) acc += partials[p];
    out[0] = acc / (float)TOT;
  }
}

// --------------------------------------------------------------------
extern "C" void kernel_launch(void* const* d_in, const int* in_sizes, int n_in,
                              void* d_out, int out_size, void* d_ws, size_t ws_size,
                              hipStream_t stream) {
  (void)in_sizes; (void)n_in; (void)out_size; (void)ws_size;
  const float* v_s0  = (const float*)d_in[0];
  const float* v_b0  = (const float*)d_in[1];
  const float* v_a0  = (const float*)d_in[2];
  const float* rbar0 = (const float*)d_in[3];
  const float* ebar0 = (const float*)d_in[4];
  const float* W_ff  = (const float*)d_in[5];
  const float* A     = (const float*)d_in[6];
  const float* inp   = (const float*)d_in[7];
  const float* noise = (const float*)d_in[8];

  float* spikes = (float*)d_out;            // [300,4,512]
  float* W_new  = spikes + TOT;             // [300,512,512]
  float* F_out  = W_new + (size_t)WTOT;     // scalar

  float* f        = (float*)d_ws;
  float* vs_init  = f;                      // TOT
  float* P        = f + (size_t)TOT;        // TOT
  float* xall     = f + 2 * (size_t)TOT;    // TOT
  float* mism     = f + 3 * (size_t)TOT;    // TOT
  float* delta    = f + 4 * (size_t)TOT;    // TOT
  float* partials = f + 5 * (size_t)TOT;    // FPART_BLOCKS
  unsigned* sync  = (unsigned*)(partials + FPART_BLOCKS);

  const int eb = (TOT + 255) / 256;
  pcsnn_init_kernel<<<eb, 256, 0, stream>>>(v_s0, noise, vs_init, sync);
  pcsnn_pre_kernel<<<eb, 256, 0, stream>>>(A, vs_init, inp, P);
  pcsnn_scan_kernel<<<SCAN_WGS, 256, 0, stream>>>(A, W_ff, v_b0, v_a0, P, vs_init,
                                                  xall, mism, delta, spikes, sync);
  pcsnn_hebb_kernel<<<(NR * 1024) / 8, 256, 0, stream>>>(W_ff, rbar0, ebar0, xall,
                                                         mism, W_new);
  pcsnn_fpart_kernel<<<FPART_BLOCKS, 256, 0, stream>>>(mism, partials);
  pcsnn_ffinal_kernel<<<1, 64, 0, stream>>>(partials, F_out);
}